// VGAE_76553497084655
// MI455X (gfx1250) — compile-verified
//
#include <hip/hip_runtime.h>

typedef __attribute__((ext_vector_type(2))) float v2f;
typedef __attribute__((ext_vector_type(8))) float v8f;

// ---------------------------------------------------------------------------
// Degree / normalization
// ---------------------------------------------------------------------------
__global__ void vgae_init_deg(float* __restrict__ deg, int n) {
    int i = blockIdx.x * blockDim.x + threadIdx.x;
    if (i < n) deg[i] = 1.0f;  // self-loop contributes 1 to every node
}

__global__ void vgae_deg_accum(const int* __restrict__ dst, float* __restrict__ deg,
                               int n_edges) {
    int e = blockIdx.x * blockDim.x + threadIdx.x;
    if (e < n_edges) atomicAdd(deg + dst[e], 1.0f);
}

__global__ void vgae_rsqrt(float* __restrict__ deg, int n) {
    int i = blockIdx.x * blockDim.x + threadIdx.x;
    if (i < n) deg[i] = rsqrtf(deg[i]);  // deg >= 1 always (self-loops)
}

// ---------------------------------------------------------------------------
// f32 WMMA GEMM: C[M,Ncols] = A[M,K] @ B[K,Ncols], one wave per 16x16 tile,
// chained V_WMMA_F32_16X16X4_F32 over K.
// ---------------------------------------------------------------------------
__global__ void vgae_gemm_wmma_f32(const float* __restrict__ A,
                                   const float* __restrict__ B,
                                   float* __restrict__ C,
                                   int Mrows, int K, int Ncols) {
    int wid  = (blockIdx.x * blockDim.x + threadIdx.x) >> 5;  // global wave id
    int lane = threadIdx.x & 31;
    int nTiles = Ncols >> 4;
    int mTiles = Mrows >> 4;
    if (wid >= mTiles * nTiles) return;  // wave-uniform exit: EXEC all-1s for WMMA
    int mTile = wid / nTiles;
    int nTile = wid - mTile * nTiles;

    int m  = lane & 15;          // A row within tile / B,C column within tile
    int kk = (lane >> 4) << 1;   // lanes 0-15 -> K=0,1 ; lanes 16-31 -> K=2,3

    const float* arow = A + (size_t)(mTile * 16 + m) * K;
    int cb = nTile * 16 + m;

    v8f acc = {};
    for (int k = 0; k < K; k += 4) {
        v2f a = *(const v2f*)(arow + k + kk);          // A[m][k+kk], A[m][k+kk+1]
        v2f b;
        b.x = B[(size_t)(k + kk)     * Ncols + cb];    // B[k+kk  ][n]
        b.y = B[(size_t)(k + kk + 1) * Ncols + cb];    // B[k+kk+1][n]
        acc = __builtin_amdgcn_wmma_f32_16x16x4_f32(
            false, a, false, b, (short)0, acc, false, false);
    }

    // D layout: VGPR r -> row r + (lane>>4)*8, col lane&15
    int rowOff = (lane >> 4) * 8;
#pragma unroll
    for (int r = 0; r < 8; ++r)
        C[(size_t)(mTile * 16 + rowOff + r) * Ncols + cb] = acc[r];
}

// ---------------------------------------------------------------------------
// acc[n,:] = dinv[n]^2 * h[n,:]   (self-loop term; initializes accumulator,
// runs BEFORE the atomic edge scatter -> no memset needed)
// ---------------------------------------------------------------------------
__global__ void vgae_selfloop_init(const float* __restrict__ h,
                                   const float* __restrict__ dinv,
                                   float* __restrict__ acc,
                                   int n_nodes, int Hq /* = H/4 */) {
    int idx = blockIdx.x * blockDim.x + threadIdx.x;  // one float4 per thread
    if (idx >= n_nodes * Hq) return;
    int node = idx / Hq;
    float s = dinv[node];
    s *= s;
    float4 v = ((const float4*)h)[idx];
    ((float4*)acc)[idx] = make_float4(v.x * s, v.y * s, v.z * s, v.w * s);
}

// ---------------------------------------------------------------------------
// Edge scatter: one wave per edge, FPL floats per lane (FPL*32 == H).
// acc[dst,:] += dinv[src]*dinv[dst] * h[src,:]
// ---------------------------------------------------------------------------
template <int FPL>
__global__ void vgae_scatter_edges(const float* __restrict__ h,
                                   float* __restrict__ acc,
                                   const int* __restrict__ src,
                                   const int* __restrict__ dst,
                                   const float* __restrict__ dinv,
                                   int n_edges) {
    int gid  = blockIdx.x * blockDim.x + threadIdx.x;
    int e    = gid >> 5;
    int lane = gid & 31;
    if (e >= n_edges) return;
    int s = src[e];
    int d = dst[e];
    float nrm = dinv[s] * dinv[d];
    const float* row = h   + (size_t)s * (FPL * 32) + lane * FPL;
    float*       out = acc + (size_t)d * (FPL * 32) + lane * FPL;
#pragma unroll
    for (int i = 0; i < FPL; ++i) atomicAdd(out + i, nrm * row[i]);
}

// ---------------------------------------------------------------------------
// In-place bias (+ optional ReLU) epilogue: acc[n,f] = act(acc[n,f] + b[f])
// ---------------------------------------------------------------------------
template <int RELU>
__global__ void vgae_bias_act(float* __restrict__ acc, const float* __restrict__ bias,
                              int total, int Hmask /* = H-1, H pow2 */) {
    int idx = blockIdx.x * blockDim.x + threadIdx.x;
    if (idx >= total) return;
    float v = acc[idx] + bias[idx & Hmask];
    if (RELU) v = fmaxf(v, 0.0f);
    acc[idx] = v;
}

// ---------------------------------------------------------------------------
// Decoder: logits[e] = dot(latent[a], latent[b]) over Z=64.
// One wave per edge, 2 floats/lane, xor-shuffle reduction (wave32).
// ---------------------------------------------------------------------------
__global__ void vgae_decode(const float* __restrict__ latent,
                            const int* __restrict__ pos,
                            const int* __restrict__ neg,
                            float* __restrict__ out, int ep) {
    int gid  = blockIdx.x * blockDim.x + threadIdx.x;
    int e    = gid >> 5;
    int lane = gid & 31;
    if (e >= 2 * ep) return;
    int a, b;
    if (e < ep) { a = pos[e];      b = pos[ep + e]; }
    else        { int j = e - ep; a = neg[j]; b = neg[ep + j]; }
    float2 va = ((const float2*)(latent + (size_t)a * 64))[lane];
    float2 vb = ((const float2*)(latent + (size_t)b * 64))[lane];
    float s = va.x * vb.x + va.y * vb.y;
#pragma unroll
    for (int off = 16; off; off >>= 1) s += __shfl_xor(s, off, 32);
    if (lane == 0) out[e] = s;
}

// ---------------------------------------------------------------------------
// Host launcher
// ---------------------------------------------------------------------------
extern "C" void kernel_launch(void* const* d_in, const int* in_sizes, int n_in,
                              void* d_out, int out_size, void* d_ws, size_t ws_size,
                              hipStream_t stream) {
    const float* x  = (const float*)d_in[0];
    const float* W1 = (const float*)d_in[1];
    const float* b1 = (const float*)d_in[2];
    const float* W2 = (const float*)d_in[3];
    const float* b2 = (const float*)d_in[4];
    const int* edge = (const int*)d_in[5];
    const int* pos  = (const int*)d_in[6];
    const int* neg  = (const int*)d_in[7];
    float* logits   = (float*)d_out;

    const int F = 128;
    const int H = in_sizes[2];       // 128
    const int Z = in_sizes[4];       // 64
    const int N = in_sizes[0] / F;   // 100000
    const int E = in_sizes[5] / 2;   // 1600000
    const int EP = in_sizes[6] / 2;  // 200000

    const int* e_src = edge;
    const int* e_dst = edge + E;

    // Workspace layout (bytes, 512-aligned). hfull region is reused for
    // lfull + acc2 after layer-1 scatter completes (stream-ordered).
    char* ws = (char*)d_ws;
    size_t off = 0;
    float* dinv = (float*)(ws + off);
    off += ((size_t)N * 4 + 511) & ~(size_t)511;
    float* hfull = (float*)(ws + off);           // [N,H] = 51.2 MB
    float* lfull = hfull;                        // [N,Z] reuses first half
    float* acc2  = hfull + (size_t)N * Z;        // [N,Z] reuses second half
    off += (size_t)N * H * 4;
    float* acc1 = (float*)(ws + off);            // [N,H] = 51.2 MB
    // total ws used ~103 MB

    const int B = 256;
    // 1) deg <- 1 (self-loops), += 1 per edge, then dinv = rsqrt(deg)
    vgae_init_deg<<<(N + B - 1) / B, B, 0, stream>>>(dinv, N);
    vgae_deg_accum<<<(E + B - 1) / B, B, 0, stream>>>(e_dst, dinv, E);
    vgae_rsqrt<<<(N + B - 1) / B, B, 0, stream>>>(dinv, N);

    // 2) hfull = x @ W1 (WMMA f32)
    {
        int tiles = (N / 16) * (H / 16);
        vgae_gemm_wmma_f32<<<(tiles * 32 + B - 1) / B, B, 0, stream>>>(x, W1, hfull, N, F, H);
    }
    // 3) acc1 = dinv^2 * hfull (self-loop init), then atomic edge scatter
    vgae_selfloop_init<<<((N * (H / 4)) + B - 1) / B, B, 0, stream>>>(hfull, dinv, acc1, N, H / 4);
    vgae_scatter_edges<4><<<((size_t)E * 32 + B - 1) / B, B, 0, stream>>>(hfull, acc1, e_src, e_dst, dinv, E);
    // 4) h = relu(acc1 + b1) in place
    vgae_bias_act<1><<<((size_t)N * H + B - 1) / B, B, 0, stream>>>(acc1, b1, N * H, H - 1);

    // 5) lfull = h @ W2 (WMMA f32)
    {
        int tiles = (N / 16) * (Z / 16);
        vgae_gemm_wmma_f32<<<(tiles * 32 + B - 1) / B, B, 0, stream>>>(acc1, W2, lfull, N, H, Z);
    }
    // 6) acc2 = dinv^2 * lfull, atomic edge scatter, + b2 (no relu)
    vgae_selfloop_init<<<((N * (Z / 4)) + B - 1) / B, B, 0, stream>>>(lfull, dinv, acc2, N, Z / 4);
    vgae_scatter_edges<2><<<((size_t)E * 32 + B - 1) / B, B, 0, stream>>>(lfull, acc2, e_src, e_dst, dinv, E);
    vgae_bias_act<0><<<((size_t)N * Z + B - 1) / B, B, 0, stream>>>(acc2, b2, N * Z, Z - 1);

    // 7) decode: 2*EP edge dot products
    vgae_decode<<<((size_t)(2 * EP) * 32 + B - 1) / B, B, 0, stream>>>(acc2, pos, neg, logits, EP);
}